// Net_19602230739296
// MI455X (gfx1250) — compile-verified
//
#include <hip/hip_runtime.h>

// ---------------------------------------------------------------------------
// NNConv + GRU message-passing net for MI455X (gfx1250), wave32 + WMMA.
//
// Algebraic transform (avoids 800 MB theta = edge_attr@nn_w materialization):
//   msg[e] = sum_a ea[e,a] * (h @ W_a)[src[e]] + (h @ reshape(nn_b))[src[e]]
// All buffers are L2-resident (~31 MB << 192 MB L2), so the run is
// launch-latency bound: fuse same-shape GEMMs to minimize launches
// (13 launches/call total).
// ---------------------------------------------------------------------------

typedef __attribute__((ext_vector_type(2))) float v2f;
typedef __attribute__((ext_vector_type(8))) float v8f;

#define NNODE 10000
#define NEDGE 50000
#define HDIM  64

// ---------------------------------------------------------------------------
// Core: one wave32 computes a 16x16 fp32 C tile with V_WMMA_F32_16X16X4_F32.
// A fragment (ISA 7.12.2): lane l holds row m0+l; K pair selected by lane
// half (hf) across the 2 VGPRs. B symmetric. K templated -> chain unrolls.
// ---------------------------------------------------------------------------
template <int K, bool RELUA, bool TRANSB>
__device__ __forceinline__ v8f wmma_tile_f32(
    const float* __restrict__ A, int lda,
    const float* __restrict__ B, int ldb,
    int m0, int col, int hf, int l)
{
    const float* __restrict__ Arow = A + (size_t)(m0 + l) * lda;
    const float* __restrict__ Bcol = TRANSB ? (B + (size_t)col * ldb) : B;
    v8f acc = {};
#pragma unroll
    for (int k = 0; k < K; k += 4) {
        const int kb = k + (hf << 1);      // half0 -> k,k+1 ; half1 -> k+2,k+3
        v2f a = *(const v2f*)(Arow + kb);  // 8B-aligned (rows are 128/256B)
        if (RELUA) { a.x = fmaxf(a.x, 0.0f); a.y = fmaxf(a.y, 0.0f); }
        v2f b;
        if (TRANSB) {
            b = *(const v2f*)(Bcol + kb);
        } else {
            b.x = B[(size_t)kb * ldb + col];
            b.y = B[(size_t)(kb + 1) * ldb + col];
        }
        acc = __builtin_amdgcn_wmma_f32_16x16x4_f32(
            false, a, false, b, (short)0, acc, false, false);
    }
    return acc;
}

// C/D layout: VGPR v -> row m0 + v + 8*hf, col n0 + lane%16.
__device__ __forceinline__ void store_tile(
    float* __restrict__ C, int ldc, int m0, int col, int hf,
    float bv, bool relu, v8f acc)
{
#pragma unroll
    for (int v = 0; v < 8; ++v) {
        const int row = m0 + v + (hf << 3);
        float r = acc[v] + bv;
        if (relu) r = fmaxf(r, 0.0f);
        C[(size_t)row * ldc + col] = r;
    }
}

// ---------------------------------------------------------------------------
// Kernel 1: h = relu(x @ lin0_w + lin0_b)   [10000,32]@[32,64]
// ---------------------------------------------------------------------------
__global__ __launch_bounds__(128) void lin0_gemm(
    const float* __restrict__ x, const float* __restrict__ w,
    const float* __restrict__ bias, float* __restrict__ h)
{
    const int wave = threadIdx.x >> 5;
    const int lane = threadIdx.x & 31;
    const int hf = lane >> 4, l = lane & 15;
    const int tile = blockIdx.x * 4 + wave;          // 2500 tiles
    if (tile >= (NNODE / 16) * (HDIM / 16)) return;
    const int tm = tile >> 2, tn = tile & 3;
    const int m0 = tm << 4, col = (tn << 4) + l;

    v8f acc = wmma_tile_f32<32, false, false>(x, 32, w, HDIM, m0, col, hf, l);
    store_tile(h, HDIM, m0, col, hf, bias[col], true, acc);
}

// ---------------------------------------------------------------------------
// Kernel 2 (fused, 6 GEMMs): for plane p in 0..5, all sharing A = h, K = 64:
//   p<4 : Y_p = h @ W_p           (W_p = nn_w + p*H*H, [64,64] row-major)
//   p=4 : Y_4 = h @ reshape(nn_b) (nn_b contribution, coefficient 1/edge)
//   p=5 : agg = h @ root_w + conv_b  (edge kernel atomically adds on top)
// plane index is wave-uniform -> EXEC stays all-ones around WMMA.
// ---------------------------------------------------------------------------
__global__ __launch_bounds__(128) void node_gemms(
    const float* __restrict__ h,
    const float* __restrict__ nn_w, const float* __restrict__ nn_b,
    const float* __restrict__ root_w, const float* __restrict__ conv_b,
    float* __restrict__ Y, float* __restrict__ agg)
{
    const int wave = threadIdx.x >> 5;
    const int lane = threadIdx.x & 31;
    const int hf = lane >> 4, l = lane & 15;
    const int tile = blockIdx.x * 4 + wave;          // 6 * 2500 tiles
    const int perPlane = (NNODE / 16) * (HDIM / 16);
    if (tile >= 6 * perPlane) return;

    const int p = tile / perPlane;
    const int t = tile - p * perPlane;
    const int tm = t >> 2, tn = t & 3;
    const int m0 = tm << 4, col = (tn << 4) + l;

    const size_t plane = (size_t)NNODE * HDIM;
    const float* B = (p < 4) ? (nn_w + (size_t)p * HDIM * HDIM)
                             : ((p == 4) ? nn_b : root_w);
    float* C = (p < 5) ? (Y + (size_t)p * plane) : agg;
    const float bv = (p == 5) ? conv_b[col] : 0.0f;

    v8f acc = wmma_tile_f32<64, false, false>(h, HDIM, B, HDIM, m0, col, hf, l);
    store_tile(C, HDIM, m0, col, hf, bv, false, acc);
}

// ---------------------------------------------------------------------------
// Kernel 3: msg[e,:] = sum_a ea[e,a]*Y_a[src[e],:] + Y_4[src[e],:]
// scatter-added into agg[dst[e],:] (global_atomic_add_f32).
// 64 threads/edge, grid-stride; prefetch next edge's gather row.
// ---------------------------------------------------------------------------
__global__ __launch_bounds__(256) void edge_msg(
    const int* __restrict__ src, const int* __restrict__ dst,
    const float* __restrict__ ea,
    const float* __restrict__ Y, float* __restrict__ agg, int E)
{
    const size_t plane = (size_t)NNODE * HDIM;
    const int t  = blockIdx.x * blockDim.x + threadIdx.x;
    const int o  = t & (HDIM - 1);
    int e        = t >> 6;
    const int es = (gridDim.x * blockDim.x) >> 6;

    for (; e < E; e += es) {
        const int en = e + es;
        if (en < E) {
            const int s2 = src[en];
            __builtin_prefetch(&Y[(size_t)s2 * HDIM + o], 0, 0);
        }
        const int s = src[e];
        const int d = dst[e];
        const float c0 = ea[e * 4 + 0];
        const float c1 = ea[e * 4 + 1];
        const float c2 = ea[e * 4 + 2];
        const float c3 = ea[e * 4 + 3];
        const size_t base = (size_t)s * HDIM + o;
        float m = Y[base + 4 * plane];
        m = fmaf(c0, Y[base + 0 * plane], m);
        m = fmaf(c1, Y[base + 1 * plane], m);
        m = fmaf(c2, Y[base + 2 * plane], m);
        m = fmaf(c3, Y[base + 3 * plane], m);
        atomicAdd(&agg[(size_t)d * HDIM + o], m);
    }
}

// ---------------------------------------------------------------------------
// Kernel 4 (fused, 2 GEMMs): [10000,64] @ [192,64]^T + bias
//   job 0: gi = relu(agg) @ w_ih^T + b_ih   (relu(agg)==m fused into A read)
//   job 1: gh = h        @ w_hh^T + b_hh
// ---------------------------------------------------------------------------
__global__ __launch_bounds__(128) void gru_gemms(
    const float* __restrict__ agg, const float* __restrict__ h,
    const float* __restrict__ w_ih, const float* __restrict__ w_hh,
    const float* __restrict__ b_ih, const float* __restrict__ b_hh,
    float* __restrict__ gi, float* __restrict__ gh)
{
    const int wave = threadIdx.x >> 5;
    const int lane = threadIdx.x & 31;
    const int hf = lane >> 4, l = lane & 15;
    const int perJob = (NNODE / 16) * (3 * HDIM / 16);   // 7500 tiles
    const int tile = blockIdx.x * 4 + wave;
    if (tile >= 2 * perJob) return;

    const int job = tile / perJob;
    const int t   = tile - job * perJob;
    const int tilesN = (3 * HDIM) / 16;                  // 12
    const int tm = t / tilesN, tn = t - tm * tilesN;
    const int m0 = tm << 4, col = (tn << 4) + l;

    v8f acc;
    if (job == 0) {
        acc = wmma_tile_f32<64, true, true>(agg, HDIM, w_ih, HDIM, m0, col, hf, l);
        store_tile(gi, 3 * HDIM, m0, col, hf, b_ih[col], false, acc);
    } else {
        acc = wmma_tile_f32<64, false, true>(h, HDIM, w_hh, HDIM, m0, col, hf, l);
        store_tile(gh, 3 * HDIM, m0, col, hf, b_hh[col], false, acc);
    }
}

// ---------------------------------------------------------------------------
// Kernel 5: GRU single-step elementwise update (in-place on h).
// ---------------------------------------------------------------------------
__global__ __launch_bounds__(256) void gru_update(
    const float* __restrict__ gi, const float* __restrict__ gh,
    float* __restrict__ h, int total)
{
    const int t = blockIdx.x * blockDim.x + threadIdx.x;
    if (t >= total) return;
    const int node = t >> 6;
    const int o    = t & (HDIM - 1);
    const float* gin = gi + (size_t)node * (3 * HDIM);
    const float* ghn = gh + (size_t)node * (3 * HDIM);

    const float sr = gin[o]        + ghn[o];
    const float sz = gin[HDIM + o] + ghn[HDIM + o];
    const float in = gin[2 * HDIM + o];
    const float hn = ghn[2 * HDIM + o];

    const float r = 1.0f / (1.0f + __expf(-sr));
    const float z = 1.0f / (1.0f + __expf(-sz));
    const float n = tanhf(in + r * hn);
    const float ho = h[t];
    h[t] = (1.0f - z) * n + z * ho;
}

// ---------------------------------------------------------------------------
// Host orchestration: graph-capturable, 13 launches per call.
// ---------------------------------------------------------------------------
extern "C" void kernel_launch(void* const* d_in, const int* in_sizes, int n_in,
                              void* d_out, int out_size, void* d_ws, size_t ws_size,
                              hipStream_t stream)
{
    (void)in_sizes; (void)n_in; (void)out_size; (void)ws_size;

    const float* x      = (const float*)d_in[0];
    const int*   eidx   = (const int*)  d_in[1];
    const float* ea     = (const float*)d_in[2];
    const float* lin0_w = (const float*)d_in[3];
    const float* lin0_b = (const float*)d_in[4];
    const float* nn_w   = (const float*)d_in[5];
    const float* nn_b   = (const float*)d_in[6];
    const float* root_w = (const float*)d_in[7];
    const float* conv_b = (const float*)d_in[8];
    const float* w_ih   = (const float*)d_in[9];
    const float* w_hh   = (const float*)d_in[10];
    const float* b_ih   = (const float*)d_in[11];
    const float* b_hh   = (const float*)d_in[12];

    const int* src = eidx;            // edge_index[0]
    const int* dst = eidx + NEDGE;    // edge_index[1]

    float* h = (float*)d_out;         // hidden state == output buffer [N,H]

    // Workspace (floats): Y[5*N*H] | agg[N*H] | gi[N*3H] | gh[N*3H] = 30.7 MB
    const size_t plane = (size_t)NNODE * HDIM;
    float* Y   = (float*)d_ws;
    float* agg = Y + 5 * plane;
    float* gi  = agg + plane;
    float* gh  = gi + (size_t)NNODE * 3 * HDIM;

    const int gLin0 = ((NNODE / 16) * (HDIM / 16) + 3) / 4;          //  625
    const int gNode = (6 * (NNODE / 16) * (HDIM / 16) + 3) / 4;      // 3750
    const int gGru  = (2 * (NNODE / 16) * (3 * HDIM / 16) + 3) / 4;  // 3750

    lin0_gemm<<<gLin0, 128, 0, stream>>>(x, lin0_w, lin0_b, h);

    for (int it = 0; it < 3; ++it) {
        node_gemms<<<gNode, 128, 0, stream>>>(h, nn_w, nn_b, root_w, conv_b,
                                              Y, agg);
        edge_msg<<<4096, 256, 0, stream>>>(src, dst, ea, Y, agg, NEDGE);
        gru_gemms<<<gGru, 128, 0, stream>>>(agg, h, w_ih, w_hh, b_ih, b_hh,
                                            gi, gh);
        gru_update<<<(NNODE * HDIM + 255) / 256, 256, 0, stream>>>(
            gi, gh, h, NNODE * HDIM);
    }
}